// KMeansClustering_19181323944384
// MI455X (gfx1250) — compile-verified
//
#include <hip/hip_runtime.h>
#include <math.h>

// K-means (64 clusters, 10 iters) + inverse-distance output for 32768x512 fp32 data.
// MI455X plan:
//  - fp32 GEMM tiles via V_WMMA_F32_16X16X4_F32 (only fp32 WMMA on CDNA5)
//  - A-tile (16 rows x 512) staged into LDS once per block via the Tensor Data
//    Mover (tensor_load_to_lds + s_wait_tensorcnt), padded to kill bank conflicts
//  - centers (128KB) stay on the global/L2 path (shared by all blocks)
//  - segment-sum via fp32 global atomics (L2 atomic units)

#define NROWS 32768
#define DIM   512
#define NC    64
#define ITERS 10

// LDS tile row stride: 512 floats + 4 pad (2-DW pad after each 256 DW from TDM
// padding). Bank base per lane = (l16*516 + 2*half) % 64 = 4*l16 + 2*half -> all
// 32 lanes of a wave land on distinct bank pairs for the b64 fragment loads.
#define TSTRIDE 516

typedef __attribute__((ext_vector_type(2))) float v2f;
typedef __attribute__((ext_vector_type(8))) float v8f;
typedef __attribute__((ext_vector_type(4))) unsigned int v4u;
typedef __attribute__((ext_vector_type(8))) int v8i;
typedef __attribute__((ext_vector_type(4))) int v4i;

#if defined(__has_builtin)
#if __has_builtin(__builtin_amdgcn_tensor_load_to_lds) && __has_builtin(__builtin_amdgcn_s_wait_tensorcnt)
#define HAVE_TDM 1
#endif
#endif
#ifndef HAVE_TDM
#define HAVE_TDM 0
#endif

__device__ __forceinline__ float atomAddF(float* p, float v) {
    return __hip_atomic_fetch_add(p, v, __ATOMIC_RELAXED, __HIP_MEMORY_SCOPE_AGENT);
}

// In-LDS column of a logical column (TDM inserts 2 DW pad after each 256 DW).
__device__ __forceinline__ int tcol(int col) { return col + ((col >= 256) ? 2 : 0); }

// Stage one 16x512 fp32 tile from global into padded LDS layout.
__device__ __forceinline__ void stage_tile(const float* __restrict__ gsrc,
                                           float* tile, int tid) {
#if HAVE_TDM
    if (tid < 32) {  // one wave issues the DMA
        unsigned long long ga = (unsigned long long)(uintptr_t)gsrc;
        unsigned lds_off = (unsigned)(uintptr_t)tile;  // low 32 bits = LDS offset
        // D# group0: count=1 | lds_addr | global_addr[56:0] | type=2
        v4u g0 = { 1u,
                   lds_off,
                   (unsigned)(ga & 0xffffffffu),
                   (unsigned)((ga >> 32) & 0x01ffffffu) | 0x80000000u };
        // D# group1:
        //  w0: data_size=2(4B)<<16 | pad_enable<<20 | pad_interval=7(256DW)<<22
        //      | pad_amount=1(2DW)<<25
        //  tensor_dim0=512 (bits 79:48), tensor_dim1=16 (bits 111:80),
        //  tile_dim0=512 (bits 127:112), tile_dim1=16 (bits 143:128),
        //  tensor_dim0_stride=512 (bits 207:160)
        v8i g1 = { (int)0x03D20000u,
                   (int)0x02000000u,   // tensor_dim0[15:0]=512 in [31:16]
                   (int)0x00100000u,   // tensor_dim1[15:0]=16  in [31:16]
                   (int)0x02000000u,   // tile_dim0=512 in [31:16]
                   16,                 // tile_dim1=16
                   512,                // tensor_dim0_stride=512
                   0, 0 };
        v4i g2 = { 0, 0, 0, 0 };
        v4i g3 = { 0, 0, 0, 0 };
#if __clang_major__ >= 23
        v8i g1b = { 0, 0, 0, 0, 0, 0, 0, 0 };
        __builtin_amdgcn_tensor_load_to_lds(g0, g1, g2, g3, g1b, 0);
#else
        __builtin_amdgcn_tensor_load_to_lds(g0, g1, g2, g3, 0);
#endif
        __builtin_amdgcn_s_wait_tensorcnt(0);
    }
    __syncthreads();
#else
    for (int idx = tid; idx < 16 * DIM; idx += 128) {
        int row = idx >> 9, col = idx & 511;
        tile[row * TSTRIDE + tcol(col)] = gsrc[idx];
    }
    __syncthreads();
#endif
}

__global__ void k_init_centers(const float* __restrict__ data, float* __restrict__ centers) {
    int i = blockIdx.x * blockDim.x + threadIdx.x;
    if (i < NC * DIM) centers[i] = data[i];
}

__global__ void k_zero(float* __restrict__ sums, float* __restrict__ counts) {
    int i = blockIdx.x * blockDim.x + threadIdx.x;
    if (i < NC * DIM) sums[i] = 0.0f;
    if (i < NC) counts[i] = 0.0f;
}

// |center_c|^2 -> cnorm[c]; one block (128 thr) per center.
__global__ void k_cnorm(const float* __restrict__ centers, float* __restrict__ cnorm) {
    int c = blockIdx.x;
    int t = threadIdx.x;
    float s = 0.0f;
    for (int j = t; j < DIM; j += 128) {
        float v = centers[c * DIM + j];
        s += v * v;
    }
    #pragma unroll
    for (int off = 16; off >= 1; off >>= 1) s += __shfl_xor(s, off, 32);
    __shared__ float red[4];
    if ((t & 31) == 0) red[t >> 5] = s;
    __syncthreads();
    if (t == 0) cnorm[c] = red[0] + red[1] + red[2] + red[3];
}

// Core WMMA tile: wave w computes dot(tile_row[m], centers[16w+n]) for 16x16.
// f32 16x16x4 fragment layout (ISA 7.12.2):
//   A: lane l holds A[l%16][k + 2*(l/16)], A[l%16][k+1 + 2*(l/16)]  (float2)
//   B: lane l holds B[k + 2*(l/16)][l%16] = centers[16w + l%16][k + 2*(l/16)..+1]
//   D: vgpr r, lane l -> row m = r + 8*(l/16), col n = l%16
__device__ __forceinline__ v8f wmma_tile_dots(const float* tile,
                                              const float* __restrict__ centers,
                                              int wave, int half, int l16) {
    const float* arow = tile + l16 * TSTRIDE + 2 * half;            // LDS (ds_load_b64)
    const float* brow = centers + (size_t)(wave * 16 + l16) * DIM + 2 * half;
    v8f acc = {};
    #pragma unroll 8
    for (int k = 0; k < 256; k += 4) {
        v2f a = *(const v2f*)(arow + k);
        v2f b = *(const v2f*)(brow + k);
        acc = __builtin_amdgcn_wmma_f32_16x16x4_f32(false, a, false, b,
                                                    (short)0, acc, false, false);
    }
    #pragma unroll 8
    for (int k = 256; k < 512; k += 4) {
        v2f a = *(const v2f*)(arow + k + 2);   // skip mid-row TDM pad
        v2f b = *(const v2f*)(brow + k);
        acc = __builtin_amdgcn_wmma_f32_16x16x4_f32(false, a, false, b,
                                                    (short)0, acc, false, false);
    }
    return acc;
}

// Assignment + segment-sum. grid = NROWS/16, block = 128 (4 wave32s).
// Each wave owns one 16-center tile; block covers all 64 centers for 16 rows.
__global__ void k_assign(const float* __restrict__ data, const float* __restrict__ centers,
                         const float* __restrict__ cnorm, float* __restrict__ sums,
                         float* __restrict__ counts) {
    const int tid  = threadIdx.x;
    const int wave = tid >> 5;
    const int lane = tid & 31;
    const int half = lane >> 4;
    const int l16  = lane & 15;
    const int row0 = blockIdx.x * 16;

    __shared__ float tile[16 * TSTRIDE];
    __shared__ float dval[16][NC + 1];   // +1 pad to spread banks
    __shared__ int   sassign[16];

    stage_tile(data + (size_t)row0 * DIM, tile, tid);

    v8f acc = wmma_tile_dots(tile, centers, wave, half, l16);

    // argmin over centers needs only |c|^2 - 2*dot (|a|^2 constant per row)
    float cn2 = cnorm[wave * 16 + l16];
    #pragma unroll
    for (int r = 0; r < 8; ++r) {
        int row = r + 8 * half;
        dval[row][wave * 16 + l16] = cn2 - 2.0f * acc[r];
    }
    __syncthreads();

    // Branchless serial argmin (strict < keeps first index, matching jnp.argmin).
    if (tid < 16) {
        float bv = dval[tid][0];
        int   bi = 0;
        #pragma unroll 8
        for (int c = 1; c < NC; ++c) {
            float v = dval[tid][c];
            bool t = v < bv;
            bv = t ? v : bv;
            bi = t ? c : bi;
        }
        sassign[tid] = bi;
        atomAddF(&counts[bi], 1.0f);
    }
    __syncthreads();

    // Segment-sum from the LDS tile into sums[assign] (L2 atomic units).
    for (int row = 0; row < 16; ++row) {
        int c = sassign[row];
        float* sr = sums + (size_t)c * DIM;
        const float* tr = tile + row * TSTRIDE;
        for (int j = tid; j < DIM; j += 128) atomAddF(&sr[j], tr[tcol(j)]);
    }
}

__global__ void k_update(const float* __restrict__ sums, const float* __restrict__ counts,
                         float* __restrict__ centers) {
    int c = blockIdx.x;
    int t = threadIdx.x;
    float cnt = counts[c];
    if (cnt > 0.0f) {
        float inv = 1.0f / cnt;
        for (int j = t; j < DIM; j += 128) centers[c * DIM + j] = sums[c * DIM + j] * inv;
    }
}

// Final: out[row][c] = 1/(sqrt(max(|a|^2 + |c|^2 - 2*dot, 0)) + 1e-8)
__global__ void k_final(const float* __restrict__ data, const float* __restrict__ centers,
                        const float* __restrict__ cnorm, float* __restrict__ out) {
    const int tid  = threadIdx.x;
    const int wave = tid >> 5;
    const int lane = tid & 31;
    const int half = lane >> 4;
    const int l16  = lane & 15;
    const int row0 = blockIdx.x * 16;

    __shared__ float tile[16 * TSTRIDE];
    __shared__ float part[16][8];
    __shared__ float a2s[16];

    stage_tile(data + (size_t)row0 * DIM, tile, tid);

    // Row norms |a|^2 from the LDS tile.
    {
        int rr = tid >> 3;   // 0..15
        int pp = tid & 7;    // 64-elem chunk (chunks 0..3 < 256, 4..7 >= 256)
        const float* tr = tile + rr * TSTRIDE + pp * 64 + ((pp >= 4) ? 2 : 0);
        float s = 0.0f;
        #pragma unroll 8
        for (int j = 0; j < 64; ++j) { float v = tr[j]; s += v * v; }
        part[rr][pp] = s;
    }
    __syncthreads();
    if (tid < 16) {
        float s = 0.0f;
        #pragma unroll
        for (int j = 0; j < 8; ++j) s += part[tid][j];
        a2s[tid] = s;
    }
    __syncthreads();

    v8f acc = wmma_tile_dots(tile, centers, wave, half, l16);

    float cn2 = cnorm[wave * 16 + l16];
    #pragma unroll
    for (int r = 0; r < 8; ++r) {
        int   row = r + 8 * half;
        float d2  = a2s[row] + cn2 - 2.0f * acc[r];
        d2 = fmaxf(d2, 0.0f);
        out[(size_t)(row0 + row) * NC + wave * 16 + l16] = 1.0f / (sqrtf(d2) + 1e-8f);
    }
}

extern "C" void kernel_launch(void* const* d_in, const int* in_sizes, int n_in,
                              void* d_out, int out_size, void* d_ws, size_t ws_size,
                              hipStream_t stream) {
    (void)in_sizes; (void)n_in; (void)out_size; (void)ws_size;
    const float* x   = (const float*)d_in[0];
    float*       out = (float*)d_out;

    // Workspace (fp32): centers[64*512] | sums[64*512] | counts[64] | cnorm[64]
    float* centers = (float*)d_ws;
    float* sums    = centers + NC * DIM;
    float* counts  = sums    + NC * DIM;
    float* cnorm   = counts  + NC;

    k_init_centers<<<(NC * DIM + 255) / 256, 256, 0, stream>>>(x, centers);

    for (int it = 0; it < ITERS; ++it) {
        k_zero<<<(NC * DIM + 255) / 256, 256, 0, stream>>>(sums, counts);
        k_cnorm<<<NC, 128, 0, stream>>>(centers, cnorm);
        k_assign<<<NROWS / 16, 128, 0, stream>>>(x, centers, cnorm, sums, counts);
        k_update<<<NC, 128, 0, stream>>>(sums, counts, centers);
    }

    k_cnorm<<<NC, 128, 0, stream>>>(centers, cnorm);
    k_final<<<NROWS / 16, 128, 0, stream>>>(x, centers, cnorm, out);
}